// AffineCurrents_73031623901593
// MI455X (gfx1250) — compile-verified
//
#include <hip/hip_runtime.h>

typedef __attribute__((ext_vector_type(2))) float v2f;
typedef __attribute__((ext_vector_type(8))) float v8f;

#define SIGMA 1.0f

// -------- workspace layout (floats) --------
// hdr[16]   : ns[9], mu[3], ut[3], pad
// set S at ws+16          : nPack[4N] | Ac[4N] | Bc[4N] | sqn[N]   (13N)
// set T at ws+16+13N      : nPack[4N] | Ac[4N] | Bc[4N] | sqn[N]   (13N)
//
// nPack[i*4 + kp*2 + c] : lane-packed normals, kp0=(nx,ny) kp1=(nz,0)
// Ac                    : kp0=(cx,cy)      kp1=(cz, 1.0)
// Bc                    : kp0=(-2cx,-2cy)  kp1=(-2cz, |c|^2)
// This matches the 32-bit 16x4 A-matrix VGPR layout: lane half selects the
// K-pair, VGPR index selects K within the pair, lane&15 selects row/col.

__global__ void prep_kernel(const float* __restrict__ srcC,
                            const float* __restrict__ affine,
                            const float* __restrict__ transl,
                            float* __restrict__ ws, int N) {
    __shared__ float red[3][256];
    int t = threadIdx.x;
    float sx = 0.f, sy = 0.f, sz = 0.f;
    for (int i = t; i < N; i += 256) {
        sx += srcC[3 * i + 0];
        sy += srcC[3 * i + 1];
        sz += srcC[3 * i + 2];
    }
    red[0][t] = sx; red[1][t] = sy; red[2][t] = sz;
    __syncthreads();
    for (int s = 128; s > 0; s >>= 1) {
        if (t < s) {
            red[0][t] += red[0][t + s];
            red[1][t] += red[1][t + s];
            red[2][t] += red[2][t + s];
        }
        __syncthreads();
    }
    if (t == 0) {
        float inv = 1.0f / (float)N;
        float mu0 = red[0][0] * inv, mu1 = red[1][0] * inv, mu2 = red[2][0] * inv;
        float a00 = affine[0], a01 = affine[1], a02 = affine[2];
        float a10 = affine[3], a11 = affine[4], a12 = affine[5];
        float a20 = affine[6], a21 = affine[7], a22 = affine[8];
        // ns = det(A) * inv(A)^T = cofactor matrix of A
        ws[0] =  (a11 * a22 - a12 * a21);
        ws[1] = -(a10 * a22 - a12 * a20);
        ws[2] =  (a10 * a21 - a11 * a20);
        ws[3] = -(a01 * a22 - a02 * a21);
        ws[4] =  (a00 * a22 - a02 * a20);
        ws[5] = -(a00 * a21 - a01 * a20);
        ws[6] =  (a01 * a12 - a02 * a11);
        ws[7] = -(a00 * a12 - a02 * a10);
        ws[8] =  (a00 * a11 - a01 * a10);
        ws[9]  = mu0; ws[10] = mu1; ws[11] = mu2;
        ws[12] = mu0 + transl[0];
        ws[13] = mu1 + transl[1];
        ws[14] = mu2 + transl[2];
        ws[15] = 0.f;
    }
}

__global__ void pack_kernel(const float* __restrict__ srcN, const float* __restrict__ srcC,
                            const float* __restrict__ tarN, const float* __restrict__ tarC,
                            const float* __restrict__ affine,
                            float* __restrict__ ws, int N, float* __restrict__ out) {
    int i = blockIdx.x * blockDim.x + threadIdx.x;
    if (i == 0) out[0] = 0.f;  // pairwise kernel atomically accumulates later
    if (i >= N) return;

    const float* ns = ws;
    const float* mu = ws + 9;
    const float* ut = ws + 12;

    float* base   = ws + 16;
    float* nPackS = base;
    float* AcS    = base + 4 * (size_t)N;
    float* BcS    = base + 8 * (size_t)N;
    float* sqnS   = base + 12 * (size_t)N;
    float* baseT  = base + 13 * (size_t)N;
    float* nPackT = baseT;
    float* AcT    = baseT + 4 * (size_t)N;
    float* BcT    = baseT + 8 * (size_t)N;
    float* sqnT   = baseT + 12 * (size_t)N;

    // ----- set S: transformed source -----
    {
        float nx = srcN[3 * i + 0], ny = srcN[3 * i + 1], nz = srcN[3 * i + 2];
        float tnx = ns[0] * nx + ns[1] * ny + ns[2] * nz;
        float tny = ns[3] * nx + ns[4] * ny + ns[5] * nz;
        float tnz = ns[6] * nx + ns[7] * ny + ns[8] * nz;
        float cx = srcC[3 * i + 0] - mu[0];
        float cy = srcC[3 * i + 1] - mu[1];
        float cz = srcC[3 * i + 2] - mu[2];
        float tcx = affine[0] * cx + affine[1] * cy + affine[2] * cz + ut[0];
        float tcy = affine[3] * cx + affine[4] * cy + affine[5] * cz + ut[1];
        float tcz = affine[6] * cx + affine[7] * cy + affine[8] * cz + ut[2];
        float sq = tcx * tcx + tcy * tcy + tcz * tcz;
        nPackS[4 * (size_t)i + 0] = tnx;
        nPackS[4 * (size_t)i + 1] = tny;
        nPackS[4 * (size_t)i + 2] = tnz;
        nPackS[4 * (size_t)i + 3] = 0.f;
        AcS[4 * (size_t)i + 0] = tcx;
        AcS[4 * (size_t)i + 1] = tcy;
        AcS[4 * (size_t)i + 2] = tcz;
        AcS[4 * (size_t)i + 3] = 1.f;
        BcS[4 * (size_t)i + 0] = -2.f * tcx;
        BcS[4 * (size_t)i + 1] = -2.f * tcy;
        BcS[4 * (size_t)i + 2] = -2.f * tcz;
        BcS[4 * (size_t)i + 3] = sq;
        sqnS[i] = sq;
    }
    // ----- set T: targets (identity) -----
    {
        float nx = tarN[3 * i + 0], ny = tarN[3 * i + 1], nz = tarN[3 * i + 2];
        float cx = tarC[3 * i + 0], cy = tarC[3 * i + 1], cz = tarC[3 * i + 2];
        float sq = cx * cx + cy * cy + cz * cz;
        nPackT[4 * (size_t)i + 0] = nx;
        nPackT[4 * (size_t)i + 1] = ny;
        nPackT[4 * (size_t)i + 2] = nz;
        nPackT[4 * (size_t)i + 3] = 0.f;
        AcT[4 * (size_t)i + 0] = cx;
        AcT[4 * (size_t)i + 1] = cy;
        AcT[4 * (size_t)i + 2] = cz;
        AcT[4 * (size_t)i + 3] = 1.f;
        BcT[4 * (size_t)i + 0] = -2.f * cx;
        BcT[4 * (size_t)i + 1] = -2.f * cy;
        BcT[4 * (size_t)i + 2] = -2.f * cz;
        BcT[4 * (size_t)i + 3] = sq;
        sqnT[i] = sq;
    }
}

// One wave owns a 16-row tile of the A side and sweeps all 16-col tiles of
// the B side. Per tile: 2x v_wmma_f32_16x16x4_f32 + 8 v_rcp_f32 epilogue.
__global__ void __launch_bounds__(256)
currents_kernel(const float* __restrict__ ws, float* __restrict__ out, int N) {
    const int lane = threadIdx.x & 31;
    const int wave = threadIdx.x >> 5;
    const int term = blockIdx.z;               // 0:SS(+1)  1:ST(-2)  2:TT(+1)
    const int i0   = (blockIdx.x * 8 + wave) * 16;

    const float* base   = ws + 16;
    const float* nPackS = base;
    const float* AcS    = base + 4 * (size_t)N;
    const float* BcS    = base + 8 * (size_t)N;
    const float* sqnS   = base + 12 * (size_t)N;
    const float* baseT  = base + 13 * (size_t)N;
    const float* nPackT = baseT;
    const float* AcT    = baseT + 4 * (size_t)N;
    const float* BcT    = baseT + 8 * (size_t)N;
    const float* sqnT   = baseT + 12 * (size_t)N;

    const float* An   = (term == 2) ? nPackT : nPackS;
    const float* Ac   = (term == 2) ? AcT    : AcS;
    const float* sqnA = (term == 2) ? sqnT   : sqnS;
    const float* Bn   = (term == 0) ? nPackS : nPackT;
    const float* Bc   = (term == 0) ? BcS    : BcT;
    const float  w    = (term == 1) ? -2.0f  : 1.0f;

    const int m  = lane & 15;   // row (A) / col (B) within tile
    const int kp = lane >> 4;   // which K-pair this half-wave holds

    // A operands: fixed for this wave
    v2f an = *(const v2f*)(An + ((size_t)(i0 + m) * 4 + kp * 2));
    v2f ac = *(const v2f*)(Ac + ((size_t)(i0 + m) * 4 + kp * 2));

    // C preload for distance WMMA: C[r] = |ca_{i0 + r + 8*kp}|^2
    v8f cD2;
#pragma unroll
    for (int r = 0; r < 8; ++r) cD2[r] = sqnA[i0 + r + 8 * kp];
    v8f cZ = {};

    const float inv_s2 = 1.0f / (SIGMA * SIGMA);
    float accL = 0.f;

    for (int j0 = 0; j0 < N; j0 += 16) {
        v2f bn = *(const v2f*)(Bn + ((size_t)(j0 + m) * 4 + kp * 2));
        v2f bc = *(const v2f*)(Bc + ((size_t)(j0 + m) * 4 + kp * 2));
        if (j0 + 16 < N) {
            __builtin_prefetch(Bn + ((size_t)(j0 + 16 + m) * 4), 0, 1);
            __builtin_prefetch(Bc + ((size_t)(j0 + 16 + m) * 4), 0, 1);
        }
        // d2_ij = |ca_i|^2 - 2 ca_i.cb_j + |cb_j|^2  (one f32 WMMA, K=4)
        v8f d2 = __builtin_amdgcn_wmma_f32_16x16x4_f32(
            false, ac, false, bc, (short)0, cD2, false, false);
        // gram_ij = na_i . nb_j                       (one f32 WMMA, K=4)
        v8f gr = __builtin_amdgcn_wmma_f32_16x16x4_f32(
            false, an, false, bn, (short)0, cZ, false, false);
#pragma unroll
        for (int r = 0; r < 8; ++r)
            accL += gr[r] * __builtin_amdgcn_rcpf(1.0f + d2[r] * inv_s2);
    }

    // wave32 reduction
    for (int off = 16; off > 0; off >>= 1)
        accL += __shfl_xor(accL, off, 32);
    if (lane == 0) atomicAdd(out, w * accL);
}

extern "C" void kernel_launch(void* const* d_in, const int* in_sizes, int n_in,
                              void* d_out, int out_size, void* d_ws, size_t ws_size,
                              hipStream_t stream) {
    const float* srcN   = (const float*)d_in[0];
    const float* srcC   = (const float*)d_in[1];
    const float* tarN   = (const float*)d_in[2];
    const float* tarC   = (const float*)d_in[3];
    const float* affine = (const float*)d_in[4];
    const float* transl = (const float*)d_in[5];
    float* out = (float*)d_out;
    float* ws  = (float*)d_ws;

    const int N = in_sizes[0] / 3;   // 8192: multiple of 128, tiles divide evenly

    prep_kernel<<<1, 256, 0, stream>>>(srcC, affine, transl, ws, N);

    const int pb = (N + 255) / 256;
    pack_kernel<<<pb, 256, 0, stream>>>(srcN, srcC, tarN, tarC, affine, ws, N, out);

    dim3 grid((N / 16) / 8, 1, 3);   // 8 waves/block, one 16-row tile per wave
    currents_kernel<<<grid, 256, 0, stream>>>(ws, out, N);
}